// GATModel_23003844838152
// MI455X (gfx1250) — compile-verified
//
#include <hip/hip_runtime.h>
#include <math.h>

#define GAT_N 50000
#define GAT_E 800000

typedef float v2f __attribute__((ext_vector_type(2)));
typedef float v8f __attribute__((ext_vector_type(8)));

// ---------------------------------------------------------------------------
// WMMA GEMM: F[N x 64] = X[N x K] @ W[K x 64]
// grid = (N/16, 64/16), block = 32 (exactly one wave32; EXEC all ones).
// Uses V_WMMA_F32_16X16X4_F32 (full fp32 precision).
//
// A-frag (16x4 f32, ISA 7.12.2): lanes 0-15 hold {K=0,K=1} for M=lane,
//                                lanes 16-31 hold {K=2,K=3} for M=lane-16.
// B-frag (4x16, mirrored):       lanes 0-15 hold {K=0,K=1} for N=lane,
//                                lanes 16-31 hold {K=2,K=3} for N=lane-16.
// C/D (16x16 f32): vgpr r -> row r (lanes 0-15) / row r+8 (lanes 16-31),
//                  col = lane % 16.
// ---------------------------------------------------------------------------
__global__ __launch_bounds__(32)
void gat_gemm_wmma_f32(const float* __restrict__ X, const float* __restrict__ W,
                       float* __restrict__ F, int K)
{
    const int lane = threadIdx.x;      // 0..31
    const int half = lane >> 4;        // 0 or 1
    const int l16  = lane & 15;
    const int row  = blockIdx.x * 16 + l16;   // A row handled by this lane
    const int col  = blockIdx.y * 16 + l16;   // B/C col handled by this lane

    v8f acc = {};
#pragma unroll 4
    for (int k0 = 0; k0 < K; k0 += 4) {
        const int ka = k0 + half * 2;
        v2f a, b;
        a.x = X[row * K + ka];
        a.y = X[row * K + ka + 1];
        b.x = W[ka * 64 + col];
        b.y = W[(ka + 1) * 64 + col];
        acc = __builtin_amdgcn_wmma_f32_16x16x4_f32(
            /*neg_a=*/false, a, /*neg_b=*/false, b,
            /*c_mod=*/(short)0, acc, /*reuse_a=*/false, /*reuse_b=*/false);
    }

    const int rbase = blockIdx.x * 16 + half * 8;
#pragma unroll
    for (int r = 0; r < 8; ++r)
        F[(rbase + r) * 64 + col] = acc[r];
}

// ---------------------------------------------------------------------------
// el[n,h] = sum_d f[n,h,d]*al[h,d] ; er likewise. One thread per (n,h).
// ---------------------------------------------------------------------------
__global__ void gat_attn_scores(const float* __restrict__ F,
                                const float* __restrict__ al,
                                const float* __restrict__ ar,
                                float* __restrict__ el, float* __restrict__ er,
                                int H, int D)
{
    int i = blockIdx.x * blockDim.x + threadIdx.x;
    if (i >= GAT_N * H) return;
    int n = i / H, h = i % H;
    const float* fp = F + (size_t)n * H * D + h * D;
    float sl = 0.f, sr = 0.f;
    for (int d = 0; d < D; ++d) {
        float v = fp[d];
        sl = fmaf(v, al[h * D + d], sl);
        sr = fmaf(v, ar[h * D + d], sr);
    }
    el[i] = sl;
    er[i] = sr;
}

__global__ void gat_init_ms(float* __restrict__ m, float* __restrict__ s, int n)
{
    int i = blockIdx.x * blockDim.x + threadIdx.x;
    if (i >= n) return;
    m[i] = -INFINITY;
    s[i] = 0.f;
}

__global__ void gat_out_init(float* __restrict__ out, const float* __restrict__ b, int C)
{
    int i = blockIdx.x * blockDim.x + threadIdx.x;
    if (i >= GAT_N * C) return;
    out[i] = b[i % C];
}

// Bitwise float atomic max (m initialized to -inf).
__device__ __forceinline__ void atomicMaxF32(float* addr, float v)
{
    if (v >= 0.f)
        atomicMax((int*)addr, __float_as_int(v));
    else
        atomicMin((unsigned int*)addr, __float_as_uint(v));
}

// e = leaky_relu(el[src]+er[dst]); segment-max into m[dst]. One thread per (edge,h).
__global__ void gat_edge_max(const int* __restrict__ src, const int* __restrict__ dst,
                             const float* __restrict__ el, const float* __restrict__ er,
                             float* __restrict__ ebuf, float* __restrict__ m, int H)
{
    int i = blockIdx.x * blockDim.x + threadIdx.x;
    if (i >= GAT_E * H) return;
    int eid = i / H, h = i % H;
    int sn = src[eid], dn = dst[eid];
    float v = el[sn * H + h] + er[dn * H + h];
    v = (v > 0.f) ? v : 0.2f * v;          // leaky_relu, NEG_SLOPE=0.2
    ebuf[i] = v;
    atomicMaxF32(&m[dn * H + h], v);
}

// ex = exp(e - m[dst]); segment-sum into s[dst]; ebuf overwritten with ex.
__global__ void gat_edge_exp(const int* __restrict__ dst,
                             float* __restrict__ ebuf, const float* __restrict__ m,
                             float* __restrict__ s, int H)
{
    int i = blockIdx.x * blockDim.x + threadIdx.x;
    if (i >= GAT_E * H) return;
    int eid = i / H, h = i % H;
    int dn = dst[eid];
    float ex = expf(ebuf[i] - m[dn * H + h]);
    ebuf[i] = ex;
    atomicAdd(&s[dn * H + h], ex);
}

// out[dst, c] += (ex/s[dst]) * f[src, c]. One thread per (edge, c), c in [0,64).
__global__ void gat_edge_agg(const int* __restrict__ src, const int* __restrict__ dst,
                             const float* __restrict__ ebuf, const float* __restrict__ s,
                             const float* __restrict__ F, float* __restrict__ out,
                             int H, int D)
{
    int i = blockIdx.x * blockDim.x + threadIdx.x;
    const int C = H * D;                         // 64
    if (i >= GAT_E * C) return;                  // 51.2M < 2^31
    int eid = i / C, c = i % C;
    int h = c / D;
    int sn = src[eid], dn = dst[eid];
    float a = ebuf[eid * H + h] / s[dn * H + h];
    atomicAdd(&out[(size_t)dn * C + c], a * F[(size_t)sn * C + c]);
}

__global__ void gat_elu(float* __restrict__ x, int n)
{
    int i = blockIdx.x * blockDim.x + threadIdx.x;
    if (i >= n) return;
    float v = x[i];
    x[i] = (v > 0.f) ? v : expm1f(v);
}

// ---------------------------------------------------------------------------
// Host-side: one GAT layer = GEMM + scores + segment softmax + aggregation.
// ---------------------------------------------------------------------------
static void run_gat_layer(const float* x, int K,
                          const float* W, const float* al, const float* ar,
                          const float* b, int H, int D,
                          const int* src, const int* dst,
                          float* fbuf, float* el, float* er, float* mb, float* sb,
                          float* ebuf, float* out, hipStream_t stream)
{
    dim3 gg(GAT_N / 16, (H * D) / 16);
    gat_gemm_wmma_f32<<<gg, 32, 0, stream>>>(x, W, fbuf, K);

    int nh = GAT_N * H;
    gat_attn_scores<<<(nh + 255) / 256, 256, 0, stream>>>(fbuf, al, ar, el, er, H, D);
    gat_init_ms<<<(nh + 255) / 256, 256, 0, stream>>>(mb, sb, nh);

    int nc = GAT_N * H * D;
    gat_out_init<<<(nc + 255) / 256, 256, 0, stream>>>(out, b, H * D);

    int eh = GAT_E * H;
    gat_edge_max<<<(eh + 255) / 256, 256, 0, stream>>>(src, dst, el, er, ebuf, mb, H);
    gat_edge_exp<<<(eh + 255) / 256, 256, 0, stream>>>(dst, ebuf, mb, sb, H);

    int ec = GAT_E * H * D;
    gat_edge_agg<<<(ec + 255) / 256, 256, 0, stream>>>(src, dst, ebuf, sb, fbuf, out, H, D);
}

extern "C" void kernel_launch(void* const* d_in, const int* in_sizes, int n_in,
                              void* d_out, int out_size, void* d_ws, size_t ws_size,
                              hipStream_t stream)
{
    (void)in_sizes; (void)n_in; (void)out_size; (void)ws_size;

    const float* h   = (const float*)d_in[0];
    const int*   src = (const int*)  d_in[1];
    const int*   dst = (const int*)  d_in[2];
    const float* W0  = (const float*)d_in[3];
    const float* al0 = (const float*)d_in[4];
    const float* ar0 = (const float*)d_in[5];
    const float* b0  = (const float*)d_in[6];
    const float* W1  = (const float*)d_in[7];
    const float* al1 = (const float*)d_in[8];
    const float* ar1 = (const float*)d_in[9];
    const float* b1  = (const float*)d_in[10];
    const float* W2  = (const float*)d_in[11];
    const float* al2 = (const float*)d_in[12];
    const float* ar2 = (const float*)d_in[13];
    const float* b2  = (const float*)d_in[14];

    // Workspace layout (floats): fbuf[N*64] xbuf[N*64] el[N*4] er[N*4]
    //                            m[N*4] s[N*4] ebuf[E*4]  (~41.6 MB)
    float* ws   = (float*)d_ws;
    float* fbuf = ws;
    float* xbuf = fbuf + (size_t)GAT_N * 64;
    float* el   = xbuf + (size_t)GAT_N * 64;
    float* er   = el + (size_t)GAT_N * 4;
    float* mb   = er + (size_t)GAT_N * 4;
    float* sb   = mb + (size_t)GAT_N * 4;
    float* ebuf = sb + (size_t)GAT_N * 4;

    // Layer 0: [N,128] -> [N,4,16]; out into xbuf (h only read by the GEMM).
    run_gat_layer(h, 128, W0, al0, ar0, b0, 4, 16, src, dst,
                  fbuf, el, er, mb, sb, ebuf, xbuf, stream);
    gat_elu<<<(GAT_N * 64 + 255) / 256, 256, 0, stream>>>(xbuf, GAT_N * 64);

    // Layer 1: xbuf -> xbuf (xbuf consumed by the GEMM before out_init rewrites it).
    run_gat_layer(xbuf, 64, W1, al1, ar1, b1, 4, 16, src, dst,
                  fbuf, el, er, mb, sb, ebuf, xbuf, stream);
    gat_elu<<<(GAT_N * 64 + 255) / 256, 256, 0, stream>>>(xbuf, GAT_N * 64);

    // Layer 2: xbuf -> d_out, H=1, D=64, no ELU.
    run_gat_layer(xbuf, 64, W2, al2, ar2, b2, 1, 64, src, dst,
                  fbuf, el, er, mb, sb, ebuf, (float*)d_out, stream);
}